// LieFucWithFixedContextWeightOptimized_30855045054564
// MI455X (gfx1250) — compile-verified
//
#include <hip/hip_runtime.h>

// ---------------------------------------------------------------------------
// B=8, D=256, S=1024. weight is [D, 1028, 1028]; strictly-lower-triangular
// 1024x1024 block per channel d.
//
// Roofline: weight lower triangle ~540 MB must stream from HBM exactly once
// -> ~23 us @ 23.3 TB/s.  FLOPs ~2.1 GF fp32 -> memory bound, so:
//   * f32 WMMA (16x16x4) consumes the stream at full rate,
//   * weight loads are non-temporal (one-shot data must not evict L2),
//   * x (8 MB) and intermediate y (8 MB) stay regular-temporal so kernel 2
//     hits L2.
// Kernel 1: per (d, i-tile) wave: C[i,b] = sum_w W[d,i,w] * x[b,d,w] via
//           v_wmma_f32_16x16x4_f32; unmasked K loop + one masked diagonal
//           tile; y = x*(1+relu(C)) -> d_ws (8 MB).
// Kernel 2: LayerNorm over D per (b,s) column -> d_out.
// ---------------------------------------------------------------------------

typedef __attribute__((ext_vector_type(2))) float v2f;
typedef __attribute__((ext_vector_type(8))) float v8f;
typedef __attribute__((ext_vector_type(4))) int   v4i;

// pointer types matching the async-copy builtin's declared parameters
typedef v4i __attribute__((address_space(1)))* g_v4i_ptr;   // global (AS1)
typedef v4i __attribute__((address_space(3)))* l_v4i_ptr;   // LDS    (AS3)

#define B_   8
#define D_   256
#define S_   1024
#define WLD  1028
#define EPS_ 1e-5f

static __device__ __forceinline__ v8f wmma_f32_4(v2f a, v2f b, v8f c) {
    return __builtin_amdgcn_wmma_f32_16x16x4_f32(
        false, a, false, b, (short)0, c, false, false);
}

__global__ __launch_bounds__(128)
void lie_mix_wmma_kernel(const float* __restrict__ x,
                         const float* __restrict__ weight,
                         float* __restrict__ y)
{
    __shared__ float xs[B_ * S_];                 // x[:, d, :]  (32 KB)

    const int d   = blockIdx.x;
    const int tid = threadIdx.x;

    // ---- stage x[:, d, :] into LDS (async-copy path when available) ----
#if __has_builtin(__builtin_amdgcn_global_load_async_to_lds_b128)
    for (int j = tid; j < (B_ * S_ / 4); j += 128) {
        const int b  = j >> 8;                    // 256 float4 per 1024-row
        const int c4 = j & 255;
        const float* src = x + (size_t)b * D_ * S_ + (size_t)d * S_
                             + (size_t)c4 * 4;
        __builtin_amdgcn_global_load_async_to_lds_b128(
            (g_v4i_ptr)(uintptr_t)src,
            (l_v4i_ptr)(xs + j * 4),
            0, 0);
    }
#if __has_builtin(__builtin_amdgcn_s_wait_asynccnt)
    __builtin_amdgcn_s_wait_asynccnt(0);
#else
    asm volatile("s_wait_asynccnt 0x0" ::: "memory");
#endif
#else
    float4* xs4 = (float4*)xs;
    for (int j = tid; j < (B_ * S_ / 4); j += 128) {
        const int b  = j >> 8;
        const int c4 = j & 255;
        xs4[j] = *(const float4*)(x + (size_t)b * D_ * S_
                                    + (size_t)d * S_ + (size_t)c4 * 4);
    }
#endif
    __syncthreads();

    const int wave    = tid >> 5;
    const int lane    = tid & 31;                 // wave32
    const int tile    = blockIdx.y + 16 * wave;   // block y: tiles y, y+16, y+32, y+48
    const int i0      = tile * 16;
    const int half    = lane >> 4;
    const int mrow    = lane & 15;
    const int row     = i0 + mrow;                // A-matrix row (i)
    const int colbase = half * 2;                 // lanes 0-15: K+{0,1}; 16-31: K+{2,3}
    const int nrow    = mrow & 7;                 // B-operand x row (cols 8-15 dead)

    const float* wrow = weight + (size_t)d * WLD * WLD + (size_t)row * WLD;
    const float* xrow = xs + nrow * S_;

    v8f acc = {0.f, 0.f, 0.f, 0.f, 0.f, 0.f, 0.f, 0.f};

    // wave-uniform scalar loop bound -> clean s_cmp loop, EXEC untouched
    const int kdiag = __builtin_amdgcn_readfirstlane(i0);

    // ---- unmasked streaming region: all K-tiles strictly below diagonal ----
    for (int k0 = 0; k0 < kdiag; k0 += 16) {
        const float* wp = wrow + k0 + colbase;
        const float* xp = xrow + k0 + colbase;
        // 4 NT weight loads in flight per iteration (streamed-once data)
        v2f w0 = __builtin_nontemporal_load((const v2f*)(wp));
        v2f w1 = __builtin_nontemporal_load((const v2f*)(wp + 4));
        v2f w2 = __builtin_nontemporal_load((const v2f*)(wp + 8));
        v2f w3 = __builtin_nontemporal_load((const v2f*)(wp + 12));
        v2f b0 = *(const v2f*)(xp);
        v2f b1 = *(const v2f*)(xp + 4);
        v2f b2 = *(const v2f*)(xp + 8);
        v2f b3 = *(const v2f*)(xp + 12);
        acc = wmma_f32_4(w0, b0, acc);
        acc = wmma_f32_4(w1, b1, acc);
        acc = wmma_f32_4(w2, b2, acc);
        acc = wmma_f32_4(w3, b3, acc);
    }

    // ---- masked diagonal 16x16 tile: keep strictly w < i ----
    {
        const int k0 = kdiag;
        #pragma unroll
        for (int j = 0; j < 4; ++j) {
            const int c0 = k0 + 4 * j + colbase;
            v2f wv = __builtin_nontemporal_load((const v2f*)(wrow + c0));
            v2f a;
            a.x = (c0     < row) ? wv.x : 0.f;
            a.y = (c0 + 1 < row) ? wv.y : 0.f;
            v2f bv = *(const v2f*)(xrow + c0);
            acc = wmma_f32_4(a, bv, acc);
        }
    }

    // C/D layout: VGPR r -> (M = i0 + r + 8*half, N = lane&15). Store b < 8.
    if (mrow < 8) {
        const int b = mrow;
        union { v8f v; float f[8]; } u;
        u.v = acc;
        #pragma unroll
        for (int r = 0; r < 8; ++r) {
            const int i = i0 + r + 8 * half;
            float c = u.f[r];
            c = c > 0.f ? c : 0.f;                // relu
            const float xv = xs[b * S_ + i];
            y[(size_t)b * D_ * S_ + (size_t)d * S_ + i] = xv * (1.f + c);
        }
    }
}

__global__ __launch_bounds__(256)
void lie_layernorm_kernel(const float* __restrict__ y,
                          const float* __restrict__ gamma,
                          const float* __restrict__ beta,
                          float* __restrict__ out)
{
    __shared__ float ssum[256];
    __shared__ float ssq[256];

    const int d  = threadIdx.x;                   // channel
    const int bs = blockIdx.x;                    // b*S + s
    const int b  = bs >> 10;
    const int s  = bs & 1023;

    const size_t idx = (size_t)b * D_ * S_ + (size_t)d * S_ + s;
    const float v = y[idx];

    ssum[d] = v;
    ssq[d]  = v * v;
    __syncthreads();

    for (int off = 128; off > 0; off >>= 1) {
        if (d < off) {
            ssum[d] += ssum[d + off];
            ssq[d]  += ssq[d + off];
        }
        __syncthreads();
    }

    const float mu  = ssum[0] * (1.0f / D_);
    const float var = ssq[0] * (1.0f / D_) - mu * mu;
    out[idx] = (v - mu) * rsqrtf(var + EPS_) * gamma[d] + beta[d];
}

extern "C" void kernel_launch(void* const* d_in, const int* in_sizes, int n_in,
                              void* d_out, int out_size, void* d_ws, size_t ws_size,
                              hipStream_t stream)
{
    const float* x      = (const float*)d_in[0];
    const float* weight = (const float*)d_in[1];
    const float* gamma  = (const float*)d_in[2];
    const float* beta   = (const float*)d_in[3];
    float* out = (float*)d_out;
    float* yws = (float*)d_ws;                    // needs B*D*S*4 = 8 MB

    dim3 g1(D_, 16);                              // 256 channels x 16 tile-groups
    lie_mix_wmma_kernel<<<g1, 128, 0, stream>>>(x, weight, yws);

    lie_layernorm_kernel<<<B_ * S_, 256, 0, stream>>>(yws, gamma, beta, out);
}